// GraphConvSparse_43112881717764
// MI455X (gfx1250) — compile-verified
//
#include <hip/hip_runtime.h>
#include <hip/hip_bf16.h>

#define N 8192
#define IN_DIM 512
#define OUT_DIM 128
#define BN 64      // columns per block in main pass
#define KSL 1024   // K range per block (8 K-slices)

typedef float v2f __attribute__((ext_vector_type(2)));
typedef float v4f __attribute__((ext_vector_type(4)));
typedef float v8f __attribute__((ext_vector_type(8)));

__device__ __forceinline__ v8f wmma_f32(v2f a, v2f b, v8f c) {
  // V_WMMA_F32_16X16X4_F32 : D = A(16x4) * B(4x16) + C(16x16), all f32
  return __builtin_amdgcn_wmma_f32_16x16x4_f32(false, a, false, b, (short)0, c,
                                               false, false);
}

// ---------------- Pass 0: zero the output accumulator region (b128) ---------
__global__ void k_zero(v4f* __restrict__ p, int n4) {
  int i = blockIdx.x * blockDim.x + threadIdx.x;
  if (i < n4) p[i] = (v4f){0.0f, 0.0f, 0.0f, 0.0f};
}

// ---------------- Pass 1: deg = rowsum(adj) + 1 ;  dis = rsqrt(deg) ---------
// One block per row; b128 nontemporal loads (adj won't be re-read in this
// order and doesn't fit L2 -> keep L2 for xw / accumulator).
__global__ void k_deg(const float* __restrict__ adj, float* __restrict__ dis) {
  __shared__ float red[256];
  const int row = blockIdx.x;
  const v4f* r = (const v4f*)(adj + (size_t)row * N);   // N/4 = 2048 vec4
  float s = 0.0f;
  for (int i = threadIdx.x; i < N / 4; i += 256) {
    const v4f v = __builtin_nontemporal_load(r + i);
    s += (v.x + v.y) + (v.z + v.w);
  }
  red[threadIdx.x] = s;
  __syncthreads();
  for (int off = 128; off > 0; off >>= 1) {
    if (threadIdx.x < off) red[threadIdx.x] += red[threadIdx.x + off];
    __syncthreads();
  }
  if (threadIdx.x == 0) dis[row] = rsqrtf(red[0] + 1.0f);
}

// ---------------- Pass 2: xw[o,n] = sum_i weight[i,o] * inputs[i,n] ---------
// grid = N/16 blocks, 256 threads (8 waves); wave w -> M tile w.
__global__ void k_xw(const float* __restrict__ weight,
                     const float* __restrict__ inputs,
                     float* __restrict__ xw) {
  const int lane = threadIdx.x & 31;
  const int wv   = threadIdx.x >> 5;
  const int lm = lane & 15, lh = lane >> 4;
  const int m0 = wv * 16;
  const int n0 = blockIdx.x * 16;
  v8f c = {};
  for (int k = 0; k < IN_DIM; k += 4) {
    const int ka = k + lh * 2;
    v2f a, b;
    a.x = weight[(ka + 0) * OUT_DIM + m0 + lm];     // A[m][k] = weight[k][m]
    a.y = weight[(ka + 1) * OUT_DIM + m0 + lm];
    b.x = inputs[(size_t)(ka + 0) * N + n0 + lm];   // B[k][n] = inputs[k][n]
    b.y = inputs[(size_t)(ka + 1) * N + n0 + lm];
    c = wmma_f32(a, b, c);
  }
#pragma unroll
  for (int r = 0; r < 8; ++r)
    xw[(size_t)(m0 + r + lh * 8) * N + n0 + lm] = c[r];
}

// ---------------- Pass 3: fused adj_norm + (xw @ adj_norm) ------------------
// grid = (N/BN, N/KSL); block = 256 threads = 8 waves, wave w -> M tile w.
__global__ void k_main(const float* __restrict__ adj,
                       const float* __restrict__ xw,
                       const float* __restrict__ dis,
                       float* __restrict__ adjn,   // adj_norm output
                       float* __restrict__ acc) {  // f32 atomic accumulator
  __shared__ float tile[16][80];   // 16 K rows x 64 cols, pad 80 (bank-skew)
  const int tid  = threadIdx.x;
  const int lane = tid & 31, wv = tid >> 5;
  const int lm = lane & 15, lh = lane >> 4;
  const int n0     = blockIdx.x * BN;
  const int kstart = blockIdx.y * KSL;
  const int m0     = wv * 16;

  // cooperative tile-load role: 16 rows x 16 threads x float4
  const int kr = tid >> 4;          // 0..15
  const int c4 = (tid & 15) * 4;    // 0..60
  const v4f disn = *(const v4f*)(dis + n0 + c4);

  v8f cacc[4] = {};                 // 4 N sub-tiles of 16

  const float* xrow = xw + (size_t)(m0 + lm) * N;

  for (int kt = 0; kt < KSL; kt += 16) {
    const int kb = kstart + kt;
    {
      const int gk = kb + kr;
      const v4f av = *(const v4f*)(adj + (size_t)gk * N + n0 + c4);
      const float dk = dis[gk];
      v4f bn;
#pragma unroll
      for (int i = 0; i < 4; ++i) {
        const int gn = n0 + c4 + i;
        const float v = av[i] + (gk == gn ? 1.0f : 0.0f);  // adj + I
        bn[i] = v * dk * disn[i];                           // D^-1/2 A D^-1/2
      }
      // stream adj_norm out (not re-read): nontemporal b128 store
      __builtin_nontemporal_store(bn, (v4f*)(adjn + (size_t)gk * N + n0 + c4));
      *(v4f*)(&tile[kr][c4]) = bn;
    }
    __syncthreads();
#pragma unroll
    for (int kk = 0; kk < 16; kk += 4) {
      const int off = kk + lh * 2;
      const v2f a = *(const v2f*)(xrow + kb + off);   // A[m][k..k+1]
#pragma unroll
      for (int nt = 0; nt < 4; ++nt) {
        v2f b;
        b.x = tile[off + 0][nt * 16 + lm];            // B[k][n]
        b.y = tile[off + 1][nt * 16 + lm];
        cacc[nt] = wmma_f32(a, b, cacc[nt]);
      }
    }
    __syncthreads();
  }

#pragma unroll
  for (int nt = 0; nt < 4; ++nt)
#pragma unroll
    for (int r = 0; r < 8; ++r) {
      const int m = m0 + r + lh * 8;
      const int n = n0 + nt * 16 + lm;
      __hip_atomic_fetch_add(&acc[(size_t)m * N + n], cacc[nt][r],
                             __ATOMIC_RELAXED, __HIP_MEMORY_SCOPE_AGENT);
    }
}

// ---------------- Pass 4: in-place ReLU on the accumulated output (b128) ----
__global__ void k_relu(v4f* __restrict__ p, int n4) {
  int i = blockIdx.x * blockDim.x + threadIdx.x;
  if (i < n4) {
    v4f v = p[i];
    v.x = fmaxf(v.x, 0.0f);
    v.y = fmaxf(v.y, 0.0f);
    v.z = fmaxf(v.z, 0.0f);
    v.w = fmaxf(v.w, 0.0f);
    p[i] = v;
  }
}

extern "C" void kernel_launch(void* const* d_in, const int* in_sizes, int n_in,
                              void* d_out, int out_size, void* d_ws, size_t ws_size,
                              hipStream_t stream) {
  const float* inputs = (const float*)d_in[0];   // [IN_DIM, N]
  const float* adj    = (const float*)d_in[1];   // [N, N]
  const float* weight = (const float*)d_in[2];   // [IN_DIM, OUT_DIM]

  float* out  = (float*)d_out;                   // outputs [OUT_DIM, N]
  float* adjn = out + (size_t)OUT_DIM * N;       // adj_norm [N, N]

  float* ws  = (float*)d_ws;
  float* dis = ws;                               // N floats
  float* xw  = ws + N;                           // OUT_DIM*N floats (~4 MB)

  const int outN4 = OUT_DIM * N / 4;

  k_zero<<<(outN4 + 255) / 256, 256, 0, stream>>>((v4f*)out, outN4);
  k_deg <<<N, 256, 0, stream>>>(adj, dis);
  k_xw  <<<N / 16, 256, 0, stream>>>(weight, inputs, xw);
  k_main<<<dim3(N / BN, N / KSL), 256, 0, stream>>>(adj, xw, dis, adjn, out);
  k_relu<<<(outN4 + 255) / 256, 256, 0, stream>>>((v4f*)out, outN4);
}